// HybridDecoderLayer_80401787781820
// MI455X (gfx1250) — compile-verified
//
#include <hip/hip_runtime.h>
#include <hip/hip_bf16.h>
#include <math.h>

// ---------------------------------------------------------------------------
// HybridDecoderLayer for MI455X (gfx1250, wave32, WMMA).
// All GEMMs use V_WMMA_F32_16X16X4_F32 (fp32 in/out -> matches fp32 reference
// exactly up to FMA ordering). Every GEMM M/N in this model is a multiple of
// 64 and every K of 32, so tiles are always full and EXEC is all-ones at every WMMA.
// ---------------------------------------------------------------------------

typedef __attribute__((ext_vector_type(2))) float v2f;
typedef __attribute__((ext_vector_type(8))) float v8f;

// Model constants (match reference)
constexpr int Lc    = 1024;   // tgt length
constexpr int LMc   = 1024;   // memory length
constexpr int Bc    = 2;      // batch
constexpr int Dc    = 512;    // model dim
constexpr int Hc    = 8;      // heads
constexpr int DHc   = 64;     // head dim
constexpr int DFFc  = 2048;
constexpr int DIc   = 1024;   // 2*D
constexpr int Nst   = 16;     // mamba state
constexpr int DTRc  = 32;     // (D+15)/16
constexpr float EPSc = 1e-5f;

// ---------------------------------------------------------------------------
// Generic batched GEMM: C[M,N] = act( A[M,K] * B[N,K]^T + bias[N] )
// A row-major (lda), B row-major (N x K, ldb), C row-major (ldc).
// act: 0 none, 1 relu, 2 exact gelu, 3 softplus.
// Block = 128 threads (4 waves). Block tile 64x64, K-chunk 32 via LDS.
// Each wave owns a 32x32 region = 2x2 WMMA tiles -> frag reuse, wmma:ds = 1:1.
// ---------------------------------------------------------------------------
__global__ void __launch_bounds__(128)
gemm_wmma(const float* __restrict__ A, long lda, long strideA,
          const float* __restrict__ Bm, long ldb, long strideB,
          float* __restrict__ C, long ldc, long strideC,
          const float* __restrict__ bias,
          int M, int N, int K, int act)
{
    // row stride 36: multiple of 4 (16B-aligned float4 stores) and
    // 36*r mod 64 distinct for r=0..15 -> conflict-free fragment reads.
    __shared__ float As[64][36];
    __shared__ float Bs[64][36];

    const int z = blockIdx.z;
    A  += (long)z * strideA;
    Bm += (long)z * strideB;
    C  += (long)z * strideC;

    const int row0 = blockIdx.y * 64;
    const int col0 = blockIdx.x * 64;
    const int tid  = threadIdx.x;
    const int wave = tid >> 5;
    const int lane = tid & 31;
    const int lanelo = lane & 15;
    const int half   = lane >> 4;
    const int wm = (wave >> 1) * 32;   // wave row offset in tile
    const int wn = (wave & 1) * 32;    // wave col offset in tile

    v8f acc00 = {}, acc01 = {}, acc10 = {}, acc11 = {};

    for (int k0 = 0; k0 < K; k0 += 32) {
        // cooperative float4 tile loads: 64x32 floats = 512 float4, 4/thread
#pragma unroll
        for (int j = 0; j < 4; j++) {
            const int u  = tid + j * 128;
            const int r  = u >> 3;
            const int cq = (u & 7) * 4;
            const float4 va = *(const float4*)&A [(long)(row0 + r) * lda + (k0 + cq)];
            const float4 vb = *(const float4*)&Bm[(long)(col0 + r) * ldb + (k0 + cq)];
            *(float4*)&As[r][cq] = va;
            *(float4*)&Bs[r][cq] = vb;
        }
        if (k0 + 32 < K) {   // prefetch next K-chunk (global_prefetch_b8)
            __builtin_prefetch(&A [(long)(row0 + (tid & 63)) * lda + (k0 + 32)], 0, 3);
            __builtin_prefetch(&Bm[(long)(col0 + (tid & 63)) * ldb + (k0 + 32)], 0, 3);
        }
        __syncthreads();
#pragma unroll
        for (int kk = 0; kk < 32; kk += 4) {
            // A 16x4 frag: lanes 0-15 M=0..15; VGPR0=K(2*half), VGPR1=K(2*half+1)
            v2f a0, a1, b0, b1;
            a0.x = As[wm + lanelo     ][kk + 2 * half + 0];
            a0.y = As[wm + lanelo     ][kk + 2 * half + 1];
            a1.x = As[wm + 16 + lanelo][kk + 2 * half + 0];
            a1.y = As[wm + 16 + lanelo][kk + 2 * half + 1];
            b0.x = Bs[wn + lanelo     ][kk + 2 * half + 0];
            b0.y = Bs[wn + lanelo     ][kk + 2 * half + 1];
            b1.x = Bs[wn + 16 + lanelo][kk + 2 * half + 0];
            b1.y = Bs[wn + 16 + lanelo][kk + 2 * half + 1];
            acc00 = __builtin_amdgcn_wmma_f32_16x16x4_f32(false, a0, false, b0, (short)0, acc00, false, false);
            acc01 = __builtin_amdgcn_wmma_f32_16x16x4_f32(false, a0, false, b1, (short)0, acc01, false, false);
            acc10 = __builtin_amdgcn_wmma_f32_16x16x4_f32(false, a1, false, b0, (short)0, acc10, false, false);
            acc11 = __builtin_amdgcn_wmma_f32_16x16x4_f32(false, a1, false, b1, (short)0, acc11, false, false);
        }
        __syncthreads();
    }

    // C/D layout: VGPR v -> row v + 8*half, lane -> col
    v8f* accs[4] = { &acc00, &acc01, &acc10, &acc11 };
#pragma unroll
    for (int ti = 0; ti < 2; ti++) {
#pragma unroll
        for (int tj = 0; tj < 2; tj++) {
            v8f acc = *accs[ti * 2 + tj];
#pragma unroll
            for (int v = 0; v < 8; v++) {
                const int r = row0 + wm + 16 * ti + v + 8 * half;
                const int c = col0 + wn + 16 * tj + lanelo;
                float val = acc[v];
                if (bias) val += bias[c];
                if (act == 1)       val = fmaxf(val, 0.0f);
                else if (act == 2)  val = 0.5f * val * (1.0f + erff(val * 0.70710678118654752f));
                else if (act == 3)  val = (val > 20.0f) ? val : log1pf(expf(val));
                C[(long)r * ldc + c] = val;
            }
        }
    }
}

static inline void gemm(hipStream_t s,
                        const float* A, long lda, long sA,
                        const float* B, long ldb, long sB,
                        float* C, long ldc, long sC,
                        const float* bias, int M, int N, int K,
                        int act, int batch)
{
    dim3 g(N / 64, M / 64, batch);
    gemm_wmma<<<g, 128, 0, s>>>(A, lda, sA, B, ldb, sB, C, ldc, sC, bias, M, N, K, act);
}

// ---------------------------------------------------------------------------
// Softmax over rows of length Lk, with pre-scale. One block per row.
// ---------------------------------------------------------------------------
__global__ void __launch_bounds__(256)
softmax_rows(float* __restrict__ S, int Lk, float scale)
{
    float* p = S + (long)blockIdx.x * Lk;
    __shared__ float red[256];
    const int tid = threadIdx.x;

    float m = -1e30f;
    for (int i = tid; i < Lk; i += 256) m = fmaxf(m, p[i] * scale);
    red[tid] = m; __syncthreads();
    for (int st = 128; st > 0; st >>= 1) {
        if (tid < st) red[tid] = fmaxf(red[tid], red[tid + st]);
        __syncthreads();
    }
    m = red[0]; __syncthreads();

    float sum = 0.0f;
    for (int i = tid; i < Lk; i += 256) {
        float e = expf(p[i] * scale - m);
        p[i] = e; sum += e;
    }
    red[tid] = sum; __syncthreads();
    for (int st = 128; st > 0; st >>= 1) {
        if (tid < st) red[tid] += red[tid + st];
        __syncthreads();
    }
    const float inv = 1.0f / red[0];
    for (int i = tid; i < Lk; i += 256) p[i] *= inv;
}

// mean over heads: out[b,q,k] = (1/H) sum_h S[b,h,q,k]
__global__ void __launch_bounds__(256)
attn_mean(const float* __restrict__ S, float* __restrict__ out, long total)
{
    long i = (long)blockIdx.x * blockDim.x + threadIdx.x;
    if (i >= total) return;
    long k = i % Lc; long t = i / Lc;
    long q = t % Lc; int  b = (int)(t / Lc);
    float s = 0.0f;
#pragma unroll
    for (int h = 0; h < Hc; h++)
        s += S[(((long)b * Hc + h) * Lc + q) * Lc + k];
    out[i] = s * (1.0f / Hc);
}

// ---------------------------------------------------------------------------
// Norms. One block (256 threads) per row of width D=512.
// ---------------------------------------------------------------------------
__global__ void __launch_bounds__(256)
rmsnorm_res(const float* __restrict__ x, const float* __restrict__ res,
            const float* __restrict__ w, float* __restrict__ out, int Dn)
{
    const long row = blockIdx.x;
    const float* px = x + row * Dn;
    const float* pr = res ? res + row * Dn : nullptr;
    float* po = out + row * Dn;
    __shared__ float red[256];
    const int tid = threadIdx.x;

    float ss = 0.0f;
    for (int i = tid; i < Dn; i += 256) {
        float v = px[i] + (pr ? pr[i] : 0.0f);
        ss += v * v;
    }
    red[tid] = ss; __syncthreads();
    for (int st = 128; st > 0; st >>= 1) {
        if (tid < st) red[tid] += red[tid + st];
        __syncthreads();
    }
    const float inv = rsqrtf(red[0] / Dn + EPSc);
    for (int i = tid; i < Dn; i += 256) {
        float v = px[i] + (pr ? pr[i] : 0.0f);
        po[i] = v * inv * w[i];
    }
}

__global__ void __launch_bounds__(256)
layernorm_res(const float* __restrict__ x, const float* __restrict__ res,
              const float* __restrict__ w, const float* __restrict__ b,
              float* __restrict__ out, int Dn)
{
    const long row = blockIdx.x;
    const float* px = x + row * Dn;
    const float* pr = res ? res + row * Dn : nullptr;
    float* po = out + row * Dn;
    __shared__ float red[256];
    const int tid = threadIdx.x;

    float s = 0.0f;
    for (int i = tid; i < Dn; i += 256) s += px[i] + (pr ? pr[i] : 0.0f);
    red[tid] = s; __syncthreads();
    for (int st = 128; st > 0; st >>= 1) {
        if (tid < st) red[tid] += red[tid + st];
        __syncthreads();
    }
    const float mean = red[0] / Dn; __syncthreads();

    float s2 = 0.0f;
    for (int i = tid; i < Dn; i += 256) {
        float v = px[i] + (pr ? pr[i] : 0.0f) - mean;
        s2 += v * v;
    }
    red[tid] = s2; __syncthreads();
    for (int st = 128; st > 0; st >>= 1) {
        if (tid < st) red[tid] += red[tid + st];
        __syncthreads();
    }
    const float inv = rsqrtf(red[0] / Dn + EPSc);
    for (int i = tid; i < Dn; i += 256) {
        float v = px[i] + (pr ? pr[i] : 0.0f);
        po[i] = (v - mean) * inv * w[i] + b[i];
    }
}

// ---------------------------------------------------------------------------
// Data-movement / elementwise kernels
// ---------------------------------------------------------------------------
// qp[((b*H+h)*Ls + l)*64 + d] = src[(l*B+b)*D + h*64 + d]
__global__ void __launch_bounds__(256)
pack_heads(const float* __restrict__ src, float* __restrict__ dst, int Ls, long total)
{
    long i = (long)blockIdx.x * blockDim.x + threadIdx.x;
    if (i >= total) return;
    int d = (int)(i & 63); long t = i >> 6;
    int l = (int)(t % Ls); t /= Ls;
    int h = (int)(t % Hc); int b = (int)(t / Hc);
    dst[i] = src[((long)l * Bc + b) * Dc + h * DHc + d];
}

// vt[((b*H+h)*64 + d)*Ls + l] = src[(l*B+b)*D + h*64 + d]  (transposed V)
__global__ void __launch_bounds__(256)
pack_heads_t(const float* __restrict__ src, float* __restrict__ dst, int Ls, long total)
{
    long i = (long)blockIdx.x * blockDim.x + threadIdx.x;
    if (i >= total) return;
    int l = (int)(i % Ls); long t = i / Ls;
    int d = (int)(t % DHc); t /= DHc;
    int h = (int)(t % Hc); int b = (int)(t / Hc);
    dst[i] = src[((long)l * Bc + b) * Dc + h * DHc + d];
}

// dst[(l*B+b)*D + h*64+d] = o[((b*H+h)*L + l)*64 + d]
__global__ void __launch_bounds__(256)
merge_heads(const float* __restrict__ o, float* __restrict__ dst, long total)
{
    long i = (long)blockIdx.x * blockDim.x + threadIdx.x;
    if (i >= total) return;
    int col = (int)(i % Dc); long r = i / Dc;
    int b = (int)(r % Bc); int l = (int)(r / Bc);
    int h = col >> 6, d = col & 63;
    dst[i] = o[(((long)b * Hc + h) * Lc + l) * DHc + d];
}

// (L,B,D) -> (B,L,D)
__global__ void __launch_bounds__(256)
lbd_to_bld(const float* __restrict__ src, float* __restrict__ dst, long total)
{
    long i = (long)blockIdx.x * blockDim.x + threadIdx.x;
    if (i >= total) return;
    int d = (int)(i % Dc); long r = i / Dc;
    int l = (int)(r % Lc); int b = (int)(r / Lc);
    dst[i] = src[((long)l * Bc + b) * Dc + d];
}

// (B,L,D) -> (L,B,D)
__global__ void __launch_bounds__(256)
bld_to_lbd(const float* __restrict__ src, float* __restrict__ dst, long total)
{
    long i = (long)blockIdx.x * blockDim.x + threadIdx.x;
    if (i >= total) return;
    int d = (int)(i % Dc); long r = i / Dc;
    int b = (int)(r % Bc); int l = (int)(r / Bc);
    dst[i] = src[((long)b * Lc + l) * Dc + d];
}

// flip along L for (B,L,W) layout
__global__ void __launch_bounds__(256)
flip_l(const float* __restrict__ src, float* __restrict__ dst, int W, long total)
{
    long i = (long)blockIdx.x * blockDim.x + threadIdx.x;
    if (i >= total) return;
    int w = (int)(i % W); long r = i / W;
    int l = (int)(r % Lc); int b = (int)(r / Lc);
    dst[((long)b * Lc + (Lc - 1 - l)) * W + w] = src[i];
}

// split xz (B*L, 2*DI) -> xm (B*L, DI), z (B*L, DI)
__global__ void __launch_bounds__(256)
split_xz(const float* __restrict__ xz, float* __restrict__ xm, float* __restrict__ z, long total)
{
    long i = (long)blockIdx.x * blockDim.x + threadIdx.x;
    if (i >= total) return;
    int c = (int)(i % DIc); long r = i / DIc;
    xm[i] = xz[r * (2 * DIc) + c];
    z[i]  = xz[r * (2 * DIc) + DIc + c];
}

// causal depthwise conv (DCONV=4, left pad 3) + SiLU, (B,L,DI) layout
__global__ void __launch_bounds__(256)
conv_silu(const float* __restrict__ xm, const float* __restrict__ w,
          const float* __restrict__ cb, float* __restrict__ out, long total)
{
    long i = (long)blockIdx.x * blockDim.x + threadIdx.x;
    if (i >= total) return;
    int c = (int)(i % DIc); long r = i / DIc;
    int l = (int)(r % Lc); int b = (int)(r / Lc);
    float acc = cb[c];
#pragma unroll
    for (int j = 0; j < 4; j++) {
        int ls = l - 3 + j;
        if (ls >= 0) acc += w[c * 4 + j] * xm[((long)b * Lc + ls) * DIc + c];
    }
    out[i] = acc / (1.0f + expf(-acc));   // silu
}

// ---------------------------------------------------------------------------
// Selective scan, lane-parallel over the 16 states:
// thread t -> (b, channel c, state n = t&15). h recurrence is per-lane;
// y = sum_n h*C via 4x __shfl_xor within the 16-lane state group.
// 32768 threads = 1024 wave32 -> good latency hiding for the serial L loop.
// ---------------------------------------------------------------------------
__global__ void __launch_bounds__(256)
mamba_scan(const float* __restrict__ delta, const float* __restrict__ xm,
           const float* __restrict__ dbc, const float* __restrict__ Alog,
           float* __restrict__ y)
{
    const int t  = blockIdx.x * 256 + threadIdx.x;
    const int n  = t & 15;
    const int ch = t >> 4;                 // 0 .. B*DI-1
    if (ch >= Bc * DIc) return;
    const int c = ch & (DIc - 1);
    const int b = ch >> 10;                // DIc == 1024
    const float Ac = -expf(Alog[c * Nst + n]);
    float h = 0.0f;
    for (int l = 0; l < Lc; l++) {
        const long r  = (long)b * Lc + l;
        const float dv = delta[r * DIc + c];   // broadcast across 16 lanes
        const float xv = xm[r * DIc + c];
        const float Bv = dbc[r * 64 + 32 + n];
        const float Cv = dbc[r * 64 + 48 + n];
        h = expf(dv * Ac) * h + dv * Bv * xv;
        float contrib = h * Cv;
#pragma unroll
        for (int m = 8; m >= 1; m >>= 1)
            contrib += __shfl_xor(contrib, m, 16);
        if (n == 0) y[r * DIc + c] = contrib;
    }
}

// y = (y + Dp[c]*xm) * silu(z)
__global__ void __launch_bounds__(256)
mamba_gate(const float* __restrict__ y, const float* __restrict__ xm,
           const float* __restrict__ z, const float* __restrict__ Dp,
           float* __restrict__ out, long total)
{
    long i = (long)blockIdx.x * blockDim.x + threadIdx.x;
    if (i >= total) return;
    int c = (int)(i % DIc);
    float zz = z[i];
    out[i] = (y[i] + Dp[c] * xm[i]) * (zz / (1.0f + expf(-zz)));
}

// bimamba outer residual: out = ff + 2*x  (reference: x + (ff + x))
__global__ void __launch_bounds__(256)
bimamba_residual(const float* __restrict__ ff, const float* __restrict__ x,
                 float* __restrict__ out, long total)
{
    long i = (long)blockIdx.x * blockDim.x + threadIdx.x;
    if (i >= total) return;
    out[i] = ff[i] + 2.0f * x[i];
}

// ---------------------------------------------------------------------------
// Host side
// ---------------------------------------------------------------------------
static inline dim3 g1(long n) { return dim3((unsigned)((n + 255) / 256)); }

struct MambaScr { float *xz, *xm, *z, *xm2, *dbc, *delta, *ybuf, *yg; };

static void run_mamba(hipStream_t s, const float* xn, float* out,
                      const float* in_w, const float* in_b,
                      const float* conv_w, const float* conv_b,
                      const float* xproj_w, const float* dt_w, const float* dt_b,
                      const float* Alog, const float* Dp,
                      const float* out_w, const float* out_b,
                      const MambaScr& ws)
{
    const int  Mrows = Bc * Lc;             // 2048
    const long nBLDI = (long)Mrows * DIc;   // 2M

    // xz = xn @ in_w^T + in_b      (2048 x 2048, K=512)
    gemm(s, xn, Dc, 0, in_w, Dc, 0, ws.xz, 2 * DIc, 0, in_b, Mrows, 2 * DIc, Dc, 0, 1);
    split_xz<<<g1(nBLDI), 256, 0, s>>>(ws.xz, ws.xm, ws.z, nBLDI);
    conv_silu<<<g1(nBLDI), 256, 0, s>>>(ws.xm, conv_w, conv_b, ws.xm2, nBLDI);
    // dbc = xm2 @ xproj_w^T        (2048 x 64, K=1024)
    gemm(s, ws.xm2, DIc, 0, xproj_w, DIc, 0, ws.dbc, 64, 0, nullptr, Mrows, 64, DIc, 0, 1);
    // delta = softplus(dt @ dt_w^T + dt_b)  (2048 x 1024, K=32; dt strided in dbc)
    gemm(s, ws.dbc, 64, 0, dt_w, DTRc, 0, ws.delta, DIc, 0, dt_b, Mrows, DIc, DTRc, 3, 1);
    // scan: one lane per (channel, state)
    mamba_scan<<<dim3((Bc * DIc * Nst) / 256), 256, 0, s>>>(ws.delta, ws.xm2, ws.dbc, Alog, ws.ybuf);
    mamba_gate<<<g1(nBLDI), 256, 0, s>>>(ws.ybuf, ws.xm2, ws.z, Dp, ws.yg, nBLDI);
    // out = yg @ out_w^T + out_b   (2048 x 512, K=1024)
    gemm(s, ws.yg, DIc, 0, out_w, DIc, 0, out, Dc, 0, out_b, Mrows, Dc, DIc, 0, 1);
}

extern "C" void kernel_launch(void* const* d_in, const int* in_sizes, int n_in,
                              void* d_out, int out_size, void* d_ws, size_t ws_size,
                              hipStream_t stream)
{
    (void)in_sizes; (void)n_in; (void)out_size; (void)ws_size;
    const float* tgt      = (const float*)d_in[0];
    const float* memory   = (const float*)d_in[1];
    const float* sa_in_w  = (const float*)d_in[2];
    const float* sa_in_b  = (const float*)d_in[3];
    const float* sa_out_w = (const float*)d_in[4];
    const float* sa_out_b = (const float*)d_in[5];
    const float* ca_in_w  = (const float*)d_in[6];
    const float* ca_in_b  = (const float*)d_in[7];
    const float* ca_out_w = (const float*)d_in[8];
    const float* ca_out_b = (const float*)d_in[9];
    const float* n1_w = (const float*)d_in[10];
    const float* n2_w = (const float*)d_in[11];
    const float* n3_w = (const float*)d_in[12];
    const float* n4_w = (const float*)d_in[13];
    const float* lin1_w = (const float*)d_in[14];
    const float* lin1_b = (const float*)d_in[15];
    const float* lin2_w = (const float*)d_in[16];
    const float* lin2_b = (const float*)d_in[17];
    const float* ln1_w = (const float*)d_in[18];
    const float* ln1_b = (const float*)d_in[19];
    const float* ln2_w = (const float*)d_in[20];
    const float* ln2_b = (const float*)d_in[21];
    const float* bff1_w = (const float*)d_in[22];
    const float* bff1_b = (const float*)d_in[23];
    const float* bff2_w = (const float*)d_in[24];
    const float* bff2_b = (const float*)d_in[25];
    const float* m_in_w   = (const float*)d_in[26];
    const float* m_in_b   = (const float*)d_in[27];
    const float* m_conv_w = (const float*)d_in[28];
    const float* m_conv_b = (const float*)d_in[29];
    const float* m_xproj_w= (const float*)d_in[30];
    const float* m_dt_w   = (const float*)d_in[31];
    const float* m_dt_b   = (const float*)d_in[32];
    const float* m_Alog   = (const float*)d_in[33];
    const float* m_D      = (const float*)d_in[34];
    const float* m_out_w  = (const float*)d_in[35];
    const float* m_out_b  = (const float*)d_in[36];

    float* out_tgt  = (float*)d_out;                          // L*B*D
    float* out_attn = (float*)d_out + (long)Lc * Bc * Dc;     // B*L*L

    // ------------------ scratch carve (floats; ~240 MB total) ---------------
    float* w = (float*)d_ws;
    size_t off = 0;
    auto alloc = [&](size_t n) { float* p = w + off; off += n; return p; };
    const size_t n1M = (size_t)Lc * Bc * Dc;        // 1,048,576
    const size_t n2M = (size_t)Bc * Lc * DIc;       // 2,097,152
    float* S    = alloc((size_t)Bc * Hc * Lc * Lc); // 16.7M
    float* ffh  = alloc((size_t)Bc * Lc * DFFc);    // 4.2M
    MambaScr ms;
    ms.xz    = alloc((size_t)Bc * Lc * 2 * DIc);    // 4.2M
    ms.xm    = alloc(n2M);
    ms.z     = alloc(n2M);
    ms.xm2   = alloc(n2M);
    ms.delta = alloc(n2M);
    ms.ybuf  = alloc(n2M);
    ms.yg    = alloc(n2M);
    ms.dbc   = alloc((size_t)Bc * Lc * 64);
    float* qbuf = alloc(n1M); float* kbuf = alloc(n1M); float* vbuf = alloc(n1M);
    float* qp   = alloc(n1M); float* kp   = alloc(n1M); float* vt   = alloc(n1M);
    float* obuf = alloc(n1M); float* omrg = alloc(n1M); float* t2   = alloc(n1M);
    float* tga  = alloc(n1M); float* tgb  = alloc(n1M);
    float* xbld = alloc(n1M); float* xn   = alloc(n1M); float* xnr  = alloc(n1M);
    float* mf   = alloc(n1M); float* mb   = alloc(n1M); float* mln  = alloc(n1M);
    float* ffo  = alloc(n1M); float* x2   = alloc(n1M);
    float* tgc  = alloc(n1M); float* tgd  = alloc(n1M);

    const int  Mr   = Lc * Bc;                 // 2048 token rows
    const long nH   = (long)Bc * Hc * Lc * DHc;
    const long nAtt = (long)Bc * Lc * Lc;

    // ======================= self-attention =================================
    gemm(stream, tgt, Dc, 0, sa_in_w,             Dc, 0, qbuf, Dc, 0, sa_in_b,          Mr, Dc, Dc, 0, 1);
    gemm(stream, tgt, Dc, 0, sa_in_w + Dc * Dc,   Dc, 0, kbuf, Dc, 0, sa_in_b + Dc,     Mr, Dc, Dc, 0, 1);
    gemm(stream, tgt, Dc, 0, sa_in_w + 2*Dc*Dc,   Dc, 0, vbuf, Dc, 0, sa_in_b + 2*Dc,   Mr, Dc, Dc, 0, 1);
    pack_heads  <<<g1(nH), 256, 0, stream>>>(qbuf, qp, Lc, nH);
    pack_heads  <<<g1(nH), 256, 0, stream>>>(kbuf, kp, Lc, nH);
    pack_heads_t<<<g1(nH), 256, 0, stream>>>(vbuf, vt, Lc, nH);
    // S = Qp @ Kp^T  (batched 16: 1024x1024, K=64)
    gemm(stream, qp, DHc, (long)Lc*DHc, kp, DHc, (long)Lc*DHc, S, Lc, (long)Lc*Lc,
         nullptr, Lc, Lc, DHc, 0, Bc * Hc);
    softmax_rows<<<Bc * Hc * Lc, 256, 0, stream>>>(S, Lc, 0.125f);   // 1/sqrt(64)
    // O = A @ V  via A @ Vt^T (batched 16: 1024x64, K=1024)
    gemm(stream, S, Lc, (long)Lc*Lc, vt, Lc, (long)Lc*DHc, obuf, DHc, (long)Lc*DHc,
         nullptr, Lc, DHc, Lc, 0, Bc * Hc);
    merge_heads<<<g1(n1M), 256, 0, stream>>>(obuf, omrg, n1M);
    gemm(stream, omrg, Dc, 0, sa_out_w, Dc, 0, t2, Dc, 0, sa_out_b, Mr, Dc, Dc, 0, 1);
    rmsnorm_res<<<Mr, 256, 0, stream>>>(tgt, t2, n1_w, tga, Dc);

    // ======================= cross-attention ================================
    gemm(stream, tga,    Dc, 0, ca_in_w,           Dc, 0, qbuf, Dc, 0, ca_in_b,        Mr, Dc, Dc, 0, 1);
    gemm(stream, memory, Dc, 0, ca_in_w + Dc*Dc,   Dc, 0, kbuf, Dc, 0, ca_in_b + Dc,   LMc*Bc, Dc, Dc, 0, 1);
    gemm(stream, memory, Dc, 0, ca_in_w + 2*Dc*Dc, Dc, 0, vbuf, Dc, 0, ca_in_b + 2*Dc, LMc*Bc, Dc, Dc, 0, 1);
    pack_heads  <<<g1(nH), 256, 0, stream>>>(qbuf, qp, Lc,  nH);
    pack_heads  <<<g1(nH), 256, 0, stream>>>(kbuf, kp, LMc, nH);
    pack_heads_t<<<g1(nH), 256, 0, stream>>>(vbuf, vt, LMc, nH);
    gemm(stream, qp, DHc, (long)Lc*DHc, kp, DHc, (long)LMc*DHc, S, LMc, (long)Lc*LMc,
         nullptr, Lc, LMc, DHc, 0, Bc * Hc);
    softmax_rows<<<Bc * Hc * Lc, 256, 0, stream>>>(S, LMc, 0.125f);
    attn_mean<<<g1(nAtt), 256, 0, stream>>>(S, out_attn, nAtt);       // 2nd output
    gemm(stream, S, LMc, (long)Lc*LMc, vt, LMc, (long)LMc*DHc, obuf, DHc, (long)Lc*DHc,
         nullptr, Lc, DHc, LMc, 0, Bc * Hc);
    merge_heads<<<g1(n1M), 256, 0, stream>>>(obuf, omrg, n1M);
    gemm(stream, omrg, Dc, 0, ca_out_w, Dc, 0, t2, Dc, 0, ca_out_b, Mr, Dc, Dc, 0, 1);
    rmsnorm_res<<<Mr, 256, 0, stream>>>(tga, t2, n2_w, tgb, Dc);

    // ======================= bimamba block ==================================
    lbd_to_bld<<<g1(n1M), 256, 0, stream>>>(tgb, xbld, n1M);
    layernorm_res<<<Mr, 256, 0, stream>>>(xbld, nullptr, ln1_w, ln1_b, xn, Dc);
    // forward direction
    run_mamba(stream, xn, mf, m_in_w, m_in_b, m_conv_w, m_conv_b,
              m_xproj_w, m_dt_w, m_dt_b, m_Alog, m_D, m_out_w, m_out_b, ms);
    // backward direction: flip, run, flip back
    flip_l<<<g1(n1M), 256, 0, stream>>>(xn, xnr, Dc, n1M);
    run_mamba(stream, xnr, tgc /*temp*/, m_in_w, m_in_b, m_conv_w, m_conv_b,
              m_xproj_w, m_dt_w, m_dt_b, m_Alog, m_D, m_out_w, m_out_b, ms);
    flip_l<<<g1(n1M), 256, 0, stream>>>(tgc, mb, Dc, n1M);
    // m = LN(mf + mb); ff = gelu(m@bff1^T+b)@bff2^T+b; out = ff + 2*x
    layernorm_res<<<Mr, 256, 0, stream>>>(mf, mb, ln2_w, ln2_b, mln, Dc);
    gemm(stream, mln, Dc, 0, bff1_w, Dc, 0, ffh, DFFc, 0, bff1_b, Mr, DFFc, Dc, 2, 1);
    gemm(stream, ffh, DFFc, 0, bff2_w, DFFc, 0, ffo, Dc, 0, bff2_b, Mr, Dc, DFFc, 0, 1);
    bimamba_residual<<<g1(n1M), 256, 0, stream>>>(ffo, xbld, x2, n1M);

    // ======================= final FFN ======================================
    bld_to_lbd<<<g1(n1M), 256, 0, stream>>>(x2, tgc, n1M);
    rmsnorm_res<<<Mr, 256, 0, stream>>>(tgc, nullptr, n3_w, tgd, Dc);
    gemm(stream, tgd, Dc, 0, lin1_w, Dc, 0, ffh, DFFc, 0, lin1_b, Mr, DFFc, Dc, 1, 1);
    gemm(stream, ffh, DFFc, 0, lin2_w, DFFc, 0, t2, Dc, 0, lin2_b, Mr, Dc, DFFc, 0, 1);
    rmsnorm_res<<<Mr, 256, 0, stream>>>(tgd, t2, n4_w, out_tgt, Dc);  // 1st output
}